// RetinaNet_39943195853062
// MI455X (gfx1250) — compile-verified
//
#include <hip/hip_runtime.h>

typedef __attribute__((ext_vector_type(2))) float v2f;
typedef __attribute__((ext_vector_type(8))) float v8f;
typedef int vec4i __attribute__((vector_size(16)));
typedef __attribute__((address_space(1))) vec4i as1_v4i;
typedef __attribute__((address_space(3))) vec4i as3_v4i;

#define DEV static __device__ __forceinline__

struct Ctrl {
  unsigned prefix;
  unsigned need;
  unsigned T;
  unsigned needEq;
  unsigned cgt;
  unsigned eqcnt;
};

__constant__ int   c_W[3]      = {168, 84, 42};
__constant__ int   c_HW[3]     = {28224, 7056, 1764};
__constant__ float c_stride[3] = {8.f, 16.f, 32.f};
__constant__ float c_basic[3]  = {32.f, 64.f, 128.f};

DEV unsigned ordkey(float f) {
  unsigned u = __float_as_uint(f);
  return (u & 0x80000000u) ? ~u : (u | 0x80000000u);
}
DEV float sigmoidf(float x) { return 1.0f / (1.0f + expf(-x)); }

// ---------------------------------------------------------------------------
// 1) Per-anchor score key: sigmoid(max logit over 80 classes), bit-ordered.
//    keys indexed s = a*HW + pos (coalesced both read & write).
// ---------------------------------------------------------------------------
__global__ void k_scoremax(const float* __restrict__ cls, unsigned* __restrict__ keys,
                           int HW, int nth) {
  int t = blockIdx.x * blockDim.x + threadIdx.x;
  if (t >= nth) return;
  int q = HW >> 2;
  int a = t / q, p4 = t - a * q;
  size_t base = (size_t)a * 80 * HW + (size_t)p4 * 4;
  float m0 = -__builtin_inff(), m1 = m0, m2 = m0, m3 = m0;
  for (int c = 0; c < 80; ++c) {
    const float4 v = *(const float4*)(cls + base + (size_t)c * HW);
    m0 = fmaxf(m0, v.x); m1 = fmaxf(m1, v.y);
    m2 = fmaxf(m2, v.z); m3 = fmaxf(m3, v.w);
  }
  uint4 k;
  k.x = ordkey(sigmoidf(m0)); k.y = ordkey(sigmoidf(m1));
  k.z = ordkey(sigmoidf(m2)); k.w = ordkey(sigmoidf(m3));
  *(uint4*)(keys + (size_t)a * HW + (size_t)p4 * 4) = k;
}

// ---------------------------------------------------------------------------
// 2) Radix-select histogram pass (counts keys matching current prefix).
// ---------------------------------------------------------------------------
__global__ void k_hist(const unsigned* __restrict__ keys, int n, const Ctrl* __restrict__ ctrl,
                       unsigned mask, int shift, int bins, unsigned* __restrict__ hist) {
  __shared__ unsigned lh[4096];
  for (int i = threadIdx.x; i < bins; i += blockDim.x) lh[i] = 0;
  __syncthreads();
  unsigned pref = ctrl->prefix & mask;
  for (int i = blockIdx.x * blockDim.x + threadIdx.x; i < n; i += gridDim.x * blockDim.x) {
    unsigned k = keys[i];
    if ((k & mask) == pref) atomicAdd(&lh[(k >> shift) & (bins - 1)], 1u);
  }
  __syncthreads();
  for (int i = threadIdx.x; i < bins; i += blockDim.x)
    if (lh[i]) atomicAdd(&hist[i], lh[i]);
}

// ---------------------------------------------------------------------------
// 3) Pick the bin where the descending cumulative count crosses `need`.
// ---------------------------------------------------------------------------
__global__ void k_selectbin(const unsigned* __restrict__ hist, int bins, int shift,
                            int round, int isFinal, Ctrl* __restrict__ ctrl) {
  __shared__ unsigned chunk[256];
  int per = bins / 256;
  unsigned s = 0;
  for (int k = 0; k < per; ++k) s += hist[threadIdx.x * per + k];
  chunk[threadIdx.x] = s;
  __syncthreads();
  if (threadIdx.x == 0) {
    unsigned need = (round == 0) ? 1000u : ctrl->need;
    unsigned cum = 0; int cb = 0;
    for (int c = 255; c >= 0; --c) {
      if (cum + chunk[c] >= need) { cb = c; break; }
      cum += chunk[c];
    }
    int b = cb * per;
    for (int bb = cb * per + per - 1; bb >= cb * per; --bb) {
      unsigned h = hist[bb];
      if (cum + h >= need) { b = bb; break; }
      cum += h;
    }
    need -= cum;                       // remaining to take from bin b (>=1)
    unsigned pfx = ctrl->prefix | ((unsigned)b << shift);
    ctrl->prefix = pfx;
    ctrl->need = need;
    if (isFinal) { ctrl->T = pfx; ctrl->needEq = need; }
  }
}

// ---------------------------------------------------------------------------
// 4) Compaction: key>T -> candidate slot; key==T -> tie list.
// ---------------------------------------------------------------------------
__global__ void k_compact(const unsigned* __restrict__ keys, int n, Ctrl* __restrict__ ctrl,
                          unsigned* __restrict__ cand, int base, unsigned* __restrict__ eqlist) {
  unsigned T = ctrl->T;
  for (int i = blockIdx.x * blockDim.x + threadIdx.x; i < n; i += gridDim.x * blockDim.x) {
    unsigned k = keys[i];
    if (k > T) {
      unsigned p = atomicAdd(&ctrl->cgt, 1u);
      if (p < 1000u) cand[base + p] = (unsigned)i;
    } else if (k == T) {
      unsigned q = atomicAdd(&ctrl->eqcnt, 1u);
      if (q < 2048u) eqlist[q] = (unsigned)i;
    }
  }
}

// ---------------------------------------------------------------------------
// 5) Resolve boundary ties: take needEq smallest reference anchor indices
//    (stable top_k tie-break). Single block, bitonic sort of 2048 u32.
// ---------------------------------------------------------------------------
__global__ void k_resolveeq(const Ctrl* __restrict__ ctrl, const unsigned* __restrict__ eqlist,
                            unsigned* __restrict__ cand, int base, int HW) {
  __shared__ unsigned sk[2048];
  unsigned n = ctrl->eqcnt; if (n > 2048u) n = 2048u;
  unsigned needEq = ctrl->needEq; if (needEq > 1000u) needEq = 1000u;
  for (int i = threadIdx.x; i < 2048; i += blockDim.x) {
    if ((unsigned)i < n) {
      unsigned s = eqlist[i];
      unsigned a = s / (unsigned)HW, pos = s - a * (unsigned)HW;
      sk[i] = pos * 9u + a;            // reference anchor index
    } else sk[i] = 0xFFFFFFFFu;
  }
  __syncthreads();
  for (unsigned k = 2; k <= 2048; k <<= 1) {
    for (unsigned j = k >> 1; j > 0; j >>= 1) {
      for (unsigned i = threadIdx.x; i < 2048; i += blockDim.x) {
        unsigned ixj = i ^ j;
        if (ixj > i) {
          bool up = ((i & k) == 0);
          unsigned x = sk[i], y = sk[ixj];
          if ((x > y) == up) { sk[i] = y; sk[ixj] = x; }
        }
      }
      __syncthreads();
    }
  }
  unsigned cgt = 1000u - needEq;
  for (unsigned i = threadIdx.x; i < needEq; i += blockDim.x) {
    unsigned aidx = sk[i];
    unsigned a = aidx % 9u, pos = aidx / 9u;
    cand[base + cgt + i] = a * (unsigned)HW + pos;
  }
}

// ---------------------------------------------------------------------------
// 6) Decode 3000 candidates: score/label recompute + box decode (f64 anchor
//    sizes like the python constant table, f32 decode math like jnp).
// ---------------------------------------------------------------------------
__global__ void k_decode(const float* __restrict__ cls0, const float* __restrict__ reg0,
                         const float* __restrict__ cls1, const float* __restrict__ reg1,
                         const float* __restrict__ cls2, const float* __restrict__ reg2,
                         const unsigned* __restrict__ cand, float* __restrict__ score,
                         int* __restrict__ label, float* __restrict__ boxes,
                         unsigned* __restrict__ tb) {
  int t = blockIdx.x * blockDim.x + threadIdx.x;
  if (t >= 3000) return;
  int lvl = t / 1000;
  const float* cls = (lvl == 0) ? cls0 : ((lvl == 1) ? cls1 : cls2);
  const float* reg = (lvl == 0) ? reg0 : ((lvl == 1) ? reg1 : reg2);
  int HW = c_HW[lvl], W = c_W[lvl];
  unsigned s = cand[t];
  int a = (int)(s / (unsigned)HW), pos = (int)(s - (unsigned)a * (unsigned)HW);

  float best = -__builtin_inff(); int lab = 0;
  for (int c = 0; c < 80; ++c) {
    float v = cls[(size_t)(a * 80 + c) * HW + pos];
    if (v > best) { best = v; lab = c; }
  }
  float sc = sigmoidf(best);
  float r0 = reg[(size_t)(a * 4 + 0) * HW + pos];
  float r1 = reg[(size_t)(a * 4 + 1) * HW + pos];
  float r2 = reg[(size_t)(a * 4 + 2) * HW + pos];
  float r3 = reg[(size_t)(a * 4 + 3) * HW + pos];

  int x = pos % W, y = pos / W;
  float cx = ((float)x + 0.5f) * c_stride[lvl];
  float cy = ((float)y + 0.5f) * c_stride[lvl];
  int ari = a / 3, si = a % 3;
  double ar  = (ari == 0) ? 0.5 : ((ari == 1) ? 1.0 : 2.0);
  double scl = (si == 0) ? 1.0 : ((si == 1) ? 1.2599210498948732 : 1.5874010519681994);
  double area = (double)c_basic[lvl] * (double)c_basic[lvl] * scl;
  float aw = (float)sqrt(ar * area);
  float ah = (float)sqrt(area / ar);

  float ox = fminf(fmaxf(r0 * aw, -32.f), 32.f);
  float oy = fminf(fmaxf(r1 * ah, -32.f), 32.f);
  float wx = aw * expf(fminf(r2, 4.135166556742356f));
  float wy = ah * expf(fminf(r3, 4.135166556742356f));
  float ctrx = cx + ox, ctry = cy + oy;
  boxes[4 * t + 0] = ctrx - 0.5f * wx;
  boxes[4 * t + 1] = ctry - 0.5f * wy;
  boxes[4 * t + 2] = ctrx + 0.5f * wx;
  boxes[4 * t + 3] = ctry + 0.5f * wy;
  score[t] = sc;
  label[t] = lab;
  unsigned aidx = (unsigned)(pos * 9 + a);
  tb[t] = ((unsigned)lvl << 30) | (aidx << 12) | (unsigned)t;  // deterministic tiebreak+slot
}

// ---------------------------------------------------------------------------
// 7) Global bitonic sort (4096 u64) matching stable argsort(-scores).
// ---------------------------------------------------------------------------
__global__ void k_sort(const float* __restrict__ score, const unsigned* __restrict__ tb,
                       unsigned* __restrict__ order) {
  __shared__ unsigned long long kk[4096];
  for (int i = threadIdx.x; i < 4096; i += blockDim.x) {
    if (i < 3000) {
      unsigned os = ordkey(score[i]);
      kk[i] = ((unsigned long long)(~os) << 32) | (unsigned long long)tb[i];
    } else kk[i] = ~0ull;
  }
  __syncthreads();
  for (unsigned k = 2; k <= 4096; k <<= 1) {
    for (unsigned j = k >> 1; j > 0; j >>= 1) {
      for (unsigned i = threadIdx.x; i < 4096; i += blockDim.x) {
        unsigned ixj = i ^ j;
        if (ixj > i) {
          bool up = ((i & k) == 0);
          unsigned long long x = kk[i], y = kk[ixj];
          if ((x > y) == up) { kk[i] = y; kk[ixj] = x; }
        }
      }
      __syncthreads();
    }
  }
  for (int i = threadIdx.x; i < 3000; i += blockDim.x)
    order[i] = (unsigned)kk[i] & 0xFFFu;
}

// ---------------------------------------------------------------------------
// 8) Gather sorted class-offset boxes + areas; ballot the valid bits.
// ---------------------------------------------------------------------------
__global__ void k_apply(const unsigned* __restrict__ order, const float* __restrict__ score,
                        const int* __restrict__ label, const float* __restrict__ boxes,
                        float* __restrict__ sbox, float* __restrict__ sarea,
                        unsigned* __restrict__ keepin) {
  int t = blockIdx.x * blockDim.x + threadIdx.x;   // 3072 threads
  float sc = -1.0f;
  float4 b = {0.f, 0.f, 0.f, 0.f};
  float ar = 0.f;
  if (t < 3000) {
    unsigned s = order[t];
    sc = score[s];
    float off = (float)label[s] * 1.0e7f;          // reference's f32 class offset
    b.x = boxes[4 * s + 0] + off;
    b.y = boxes[4 * s + 1] + off;
    b.z = boxes[4 * s + 2] + off;
    b.w = boxes[4 * s + 3] + off;
    ar = (b.z - b.x) * (b.w - b.y);
  }
  ((float4*)sbox)[t] = b;
  sarea[t] = ar;
  unsigned bal = __builtin_amdgcn_ballot_w32(sc >= 0.05f);
  if ((threadIdx.x & 31) == 0) keepin[t >> 5] = bal;
}

// ---------------------------------------------------------------------------
// 9) Suppression bitmask: one wave per 16x16 tile. v_wmma_f32_16x16x4_f32
//    supplies the area_i tile (area_row x 1; area_col meets A's zero lanes, so
//    adds nothing) so the matrix pipe co-executes with VALU min/max.
// ---------------------------------------------------------------------------
__global__ void k_mask(const float* __restrict__ sbox, const float* __restrict__ sarea,
                       unsigned short* __restrict__ mask16) {
  __shared__ float4 lrb[8][16];
  __shared__ float4 lcb[8][16];
  __shared__ float  lra[8][16];
  __shared__ float  lca[8][16];
  int gt = blockIdx.x * blockDim.x + threadIdx.x;
  int wave = gt >> 5;
  int lane = threadIdx.x & 31;
  int wi = threadIdx.x >> 5;
  int rt = wave / 192, ct = wave - rt * 192;
  if (rt >= 188) return;
  if (ct < rt) return;                 // tile entirely at j<=i: memset-zero covers it
  int I0 = rt * 16, J0 = ct * 16;

  if (lane < 16) {
    int r = I0 + lane;
    lrb[wi][lane] = ((const float4*)sbox)[r];
    lra[wi][lane] = sarea[r];
  } else {
    int c = J0 + lane - 16;
    lcb[wi][lane - 16] = ((const float4*)sbox)[c];
    lca[wi][lane - 16] = sarea[c];
  }

  // A (16x4): A[m][0]=area_row[m], A[m][1]=1 ; B (4x16): B[0][n]=1, B[1][n]=0, B[2][n]=area_col[n] (A[m][2]=0)
  v2f A, B;
  A.x = (lane < 16) ? lra[wi][lane] : 0.f;
  A.y = (lane < 16) ? 1.f : 0.f;
  B.x = (lane < 16) ? 1.f : lca[wi][lane - 16];
  B.y = 0.f;
  v8f C = {0.f, 0.f, 0.f, 0.f, 0.f, 0.f, 0.f, 0.f};
  C = __builtin_amdgcn_wmma_f32_16x16x4_f32(false, A, false, B, (short)0, C, false, false);

  int n = lane & 15;
  int j = J0 + n;
  float4 cb = lcb[wi][n];
  for (int v = 0; v < 8; ++v) {
    int m = v + ((lane >= 16) ? 8 : 0);
    int i = I0 + m;
    float4 rb = lrb[wi][m];
    float ltx = fmaxf(rb.x, cb.x), lty = fmaxf(rb.y, cb.y);
    float rbx = fminf(rb.z, cb.z), rby = fminf(rb.w, cb.w);
    float wx = fmaxf(rbx - ltx, 1e-28f), wy = fmaxf(rby - lty, 1e-28f);
    float inter = wx * wy;
    float den = C[v] - inter + 1e-14f;           // area_i - inter + eps (C holds area_i only)
    float iou = inter / den;
    bool pred = (iou > 0.6f) && (j > i) && (j < 3000) && (i < 3000);
    unsigned bal = __builtin_amdgcn_ballot_w32(pred);
    if (lane == 0) {
      mask16[(size_t)(I0 + v) * 192 + ct]     = (unsigned short)(bal & 0xFFFFu);
      mask16[(size_t)(I0 + v + 8) * 192 + ct] = (unsigned short)(bal >> 16);
    }
  }
}

// ---------------------------------------------------------------------------
// 10) Serial NMS scan: single wave, keep-bitset in LDS, mask rows streamed
//     through LDS with double-buffered async loads (ASYNCcnt).
// ---------------------------------------------------------------------------
DEV void async_row128(const void* g, void* l) {
  __builtin_amdgcn_global_load_async_to_lds_b128((as1_v4i*)g, (as3_v4i*)l, 0, 0);
}

__global__ __launch_bounds__(32) void k_nms(const unsigned short* __restrict__ mask16,
                                            const unsigned* __restrict__ keepin,
                                            unsigned* __restrict__ keepout) {
  __shared__ unsigned keep[96];
  __shared__ __align__(16) unsigned rowbuf[2][96];
  int lane = threadIdx.x;
  keep[lane]      = keepin[lane];
  keep[lane + 32] = keepin[lane + 32];
  keep[lane + 64] = keepin[lane + 64];
  const unsigned char* mbase = (const unsigned char*)mask16;  // row pitch 384B = 96 dwords

  if (lane < 24) async_row128(mbase + lane * 16, &rowbuf[0][lane * 4]);
  for (int i = 0; i < 3000; ++i) {
    int cur = i & 1;
    if (i + 1 < 3000) {
      if (lane < 24)
        async_row128(mbase + (size_t)(i + 1) * 384 + lane * 16, &rowbuf[cur ^ 1][lane * 4]);
      __builtin_amdgcn_s_wait_asynccnt(1);       // row i landed, row i+1 in flight
    } else {
      __builtin_amdgcn_s_wait_asynccnt(0);
    }
    asm volatile("" ::: "memory");
    unsigned bit = (keep[i >> 5] >> (i & 31)) & 1u;
    if (bit) {
      keep[lane]      &= ~rowbuf[cur][lane];
      keep[lane + 32] &= ~rowbuf[cur][lane + 32];
      keep[lane + 64] &= ~rowbuf[cur][lane + 64];
    }
    asm volatile("" ::: "memory");
  }
  keepout[lane]      = keep[lane];
  keepout[lane + 32] = keep[lane + 32];
  keepout[lane + 64] = keep[lane + 64];
}

// ---------------------------------------------------------------------------
// 11) Final outputs: [boxes 12000 | scores 3000 | labels 3000 | keep 3000]
// ---------------------------------------------------------------------------
__global__ void k_final(const unsigned* __restrict__ order, const float* __restrict__ score,
                        const int* __restrict__ label, const float* __restrict__ boxes,
                        const unsigned* __restrict__ keepout, const int* __restrict__ imh,
                        const int* __restrict__ imw, float* __restrict__ out) {
  int t = blockIdx.x * blockDim.x + threadIdx.x;
  if (t >= 3000) return;
  unsigned s = order[t];
  unsigned k = (keepout[t >> 5] >> (t & 31)) & 1u;
  float fw = (float)imw[0], fh = (float)imh[0];
  float b0 = fminf(fmaxf(boxes[4 * s + 0] / fw, 0.f), 1.f);
  float b1 = fminf(fmaxf(boxes[4 * s + 1] / fh, 0.f), 1.f);
  float b2 = fminf(fmaxf(boxes[4 * s + 2] / fw, 0.f), 1.f);
  float b3 = fminf(fmaxf(boxes[4 * s + 3] / fh, 0.f), 1.f);
  if (!k) { b0 = 0.f; b1 = 0.f; b2 = 0.f; b3 = 0.f; }
  out[4 * t + 0] = b0; out[4 * t + 1] = b1; out[4 * t + 2] = b2; out[4 * t + 3] = b3;
  out[12000 + t] = k ? score[s] : 0.f;
  out[15000 + t] = (float)label[s];
  out[18000 + t] = k ? 1.f : 0.f;
}

// ---------------------------------------------------------------------------
extern "C" void kernel_launch(void* const* d_in, const int* in_sizes, int n_in,
                              void* d_out, int out_size, void* d_ws, size_t ws_size,
                              hipStream_t stream) {
  (void)in_sizes; (void)n_in; (void)out_size; (void)ws_size;
  const float* cls[3] = {(const float*)d_in[0], (const float*)d_in[2], (const float*)d_in[4]};
  const float* reg[3] = {(const float*)d_in[1], (const float*)d_in[3], (const float*)d_in[5]};
  const int* imh = (const int*)d_in[6];
  const int* imw = (const int*)d_in[7];

  char* w = (char*)d_ws;
  unsigned*        keys    = (unsigned*)(w + 0);          // u32[254016]
  unsigned*        hist    = (unsigned*)(w + 1016064);    // u32[4096]
  Ctrl*            ctrl    = (Ctrl*)(w + 1032448);
  unsigned*        eqlist  = (unsigned*)(w + 1032704);    // u32[2048]
  unsigned*        cand    = (unsigned*)(w + 1040896);    // u32[3000]
  float*           score   = (float*)(w + 1052928);
  int*             label   = (int*)(w + 1064960);
  unsigned*        tb      = (unsigned*)(w + 1076992);
  float*           boxes   = (float*)(w + 1089024);       // f32[3000*4]
  unsigned*        order   = (unsigned*)(w + 1137152);
  float*           sbox    = (float*)(w + 1149184);       // f32[3072*4]
  float*           sarea   = (float*)(w + 1198336);       // f32[3072]
  unsigned*        keepin  = (unsigned*)(w + 1210624);    // u32[96]
  unsigned*        keepout = (unsigned*)(w + 1211136);    // u32[96]
  unsigned short*  mask16  = (unsigned short*)(w + 1211648); // u16[3008*192]

  const int HWs[3] = {28224, 7056, 1764};

  for (int lvl = 0; lvl < 3; ++lvl) {
    int HW = HWs[lvl], HWA = 9 * HW;
    (void)hipMemsetAsync(ctrl, 0, sizeof(Ctrl), stream);
    int nth = 9 * HW / 4;
    k_scoremax<<<(nth + 255) / 256, 256, 0, stream>>>(cls[lvl], keys, HW, nth);
    int hb = (HWA + 255) / 256;

    (void)hipMemsetAsync(hist, 0, 4096 * sizeof(unsigned), stream);
    k_hist<<<hb, 256, 0, stream>>>(keys, HWA, ctrl, 0u, 20, 4096, hist);
    k_selectbin<<<1, 256, 0, stream>>>(hist, 4096, 20, 0, 0, ctrl);

    (void)hipMemsetAsync(hist, 0, 4096 * sizeof(unsigned), stream);
    k_hist<<<hb, 256, 0, stream>>>(keys, HWA, ctrl, 0xFFF00000u, 8, 4096, hist);
    k_selectbin<<<1, 256, 0, stream>>>(hist, 4096, 8, 1, 0, ctrl);

    (void)hipMemsetAsync(hist, 0, 256 * sizeof(unsigned), stream);
    k_hist<<<hb, 256, 0, stream>>>(keys, HWA, ctrl, 0xFFFFFF00u, 0, 256, hist);
    k_selectbin<<<1, 256, 0, stream>>>(hist, 256, 0, 2, 1, ctrl);

    k_compact<<<hb, 256, 0, stream>>>(keys, HWA, ctrl, cand, lvl * 1000, eqlist);
    k_resolveeq<<<1, 256, 0, stream>>>(ctrl, eqlist, cand, lvl * 1000, HW);
  }

  k_decode<<<12, 256, 0, stream>>>(cls[0], reg[0], cls[1], reg[1], cls[2], reg[2],
                                   cand, score, label, boxes, tb);
  k_sort<<<1, 512, 0, stream>>>(score, tb, order);
  k_apply<<<12, 256, 0, stream>>>(order, score, label, boxes, sbox, sarea, keepin);
  (void)hipMemsetAsync(mask16, 0, (size_t)3008 * 192 * sizeof(unsigned short), stream);
  k_mask<<<4512, 256, 0, stream>>>(sbox, sarea, mask16);
  k_nms<<<1, 32, 0, stream>>>(mask16, keepin, keepout);
  k_final<<<12, 256, 0, stream>>>(order, score, label, boxes, keepout, imh, imw, (float*)d_out);
}